// SSM_cha_Last_36386962932306
// MI455X (gfx1250) — compile-verified
//
#include <hip/hip_runtime.h>
#include <math.h>

typedef __attribute__((ext_vector_type(2))) float v2f;
typedef __attribute__((ext_vector_type(4))) float v4f;
typedef __attribute__((ext_vector_type(8))) float v8f;

#define LTOT   4096
#define DIN    256
#define BATCH  8
#define NROWS  (BATCH * LTOT)   // 32768
#define NC_DBL 272              // 256 dt cols + 4 (B0,B1,C0,C1) + 12 zero pad -> 17 tiles
#define NCHUNK 32               // scan chunks per batch
#define CLEN   (LTOT / NCHUNK)  // 128 steps per chunk
#define CHTOT  (BATCH * NCHUNK * DIN)

__device__ __forceinline__ v8f wmma_f32(v2f a, v2f b, v8f c) {
    // 8 args: (neg_a, A, neg_b, B, c_mod, C, reuse_a, reuse_b)
    return __builtin_amdgcn_wmma_f32_16x16x4_f32(false, a, false, b, (short)0, c, false, false);
}

__device__ __forceinline__ void zero8(v8f& v) {
#pragma unroll
    for (int i = 0; i < 8; ++i) v[i] = 0.0f;
}

__device__ __forceinline__ float softplus_f(float x) {
    return (x > 20.0f) ? x : __logf(1.0f + __expf(x));
}

__device__ __forceinline__ float silu_f(float x) {
    return x / (1.0f + __expf(-x));
}

// ---------------------------------------------------------------------------
// Setup: Wcat[k, n] (256 x 272):
//   n <  256 : (W_x[:, :16] @ W_dt)[k, n]   (fused dt projection)
//   256..259 : W_x[k, 16 + (n-256)]         (B0, B1, C0, C1 columns)
//   else     : 0 (pad to 17 WMMA tiles)
// ---------------------------------------------------------------------------
__global__ void build_wcat_kernel(const float* __restrict__ W_x,
                                  const float* __restrict__ W_dt,
                                  float* __restrict__ Wcat) {
    int idx = blockIdx.x * blockDim.x + threadIdx.x;
    if (idx >= DIN * NC_DBL) return;
    int k = idx / NC_DBL;
    int n = idx % NC_DBL;
    float v = 0.0f;
    if (n < DIN) {
#pragma unroll
        for (int r = 0; r < 16; ++r)
            v = fmaf(W_x[k * 20 + r], W_dt[r * DIN + n], v);
    } else if (n < DIN + 4) {
        v = W_x[k * 20 + 16 + (n - DIN)];
    }
    Wcat[idx] = v;
}

// ---------------------------------------------------------------------------
// GEMM-A: xz = u @ W_in.  Block = 4 waves = 4 M-tiles sharing one 64-col
// W_in strip staged through LDS (two 128-row stages of 32 KB).  A reads come
// from the NCHW input, column-contiguous (lanes 0..15 = consecutive l).
// ---------------------------------------------------------------------------
__global__ void gemm_in_kernel(const float* __restrict__ x,
                               const float* __restrict__ Win,
                               float* __restrict__ xi_raw,
                               float* __restrict__ zbuf) {
    __shared__ float wlds[128 * 64];           // 32 KB weight stage
    const int tid  = threadIdx.x;
    const int wave = tid >> 5;
    const int lane = tid & 31;
    const int row0 = blockIdx.x * 64 + wave * 16;
    const int ns   = blockIdx.y;               // N strip of 64
    const int b    = row0 / LTOT;
    const int l0   = row0 % LTOT;
    const int m    = lane & 15;                // A frag: M = lane%16
    const int kh   = lane >> 4;                // K half: lanes>=16 hold K=2..3

    v8f acc[4];
#pragma unroll
    for (int t = 0; t < 4; ++t) zero8(acc[t]);

    const float* xb = x + (size_t)b * DIN * LTOT + l0 + m;
    const float* wg = Win + (size_t)ns * 64;

    for (int ks = 0; ks < DIN; ks += 128) {
        // cooperative stage of Win[ks..ks+128) x strip (v4f, coalesced)
        for (int i = tid; i < 128 * 16; i += 128) {
            const int row = i >> 4;
            const int c4  = (i & 15) * 4;
            *reinterpret_cast<v4f*>(&wlds[row * 64 + c4]) =
                *reinterpret_cast<const v4f*>(&wg[(size_t)(ks + row) * 512 + c4]);
        }
        __syncthreads();
        for (int k0 = 0; k0 < 128; k0 += 4) {
            const int kl = k0 + 2 * kh;        // local K row for this lane
            v2f a;
            a.x = xb[(size_t)(ks + kl + 0) * LTOT];
            a.y = xb[(size_t)(ks + kl + 1) * LTOT];
#pragma unroll
            for (int t = 0; t < 4; ++t) {
                v2f bv;
                bv.x = wlds[kl * 64 + t * 16 + m];
                bv.y = wlds[(kl + 1) * 64 + t * 16 + m];
                acc[t] = wmma_f32(a, bv, acc[t]);
            }
        }
        __syncthreads();
    }

    // block-uniform destination select (ns<4 -> xi half, else z half)
    float* dstbuf;
    int colbase;
    if (ns < 4) { dstbuf = xi_raw; colbase = ns * 64; }
    else        { dstbuf = zbuf;   colbase = ns * 64 - 256; }
    const int mo = kh * 8;                     // lanes 16..31 hold rows M=r+8
#pragma unroll
    for (int t = 0; t < 4; ++t) {
        const int col = colbase + t * 16 + m;
#pragma unroll
        for (int r = 0; r < 8; ++r)
            dstbuf[(size_t)(row0 + mo + r) * DIN + col] = acc[t][r];
    }
}

// ---------------------------------------------------------------------------
// Depthwise causal conv (K=4) + bias + SiLU.  lanes = d -> fully coalesced.
// ---------------------------------------------------------------------------
__global__ void conv_silu_kernel(const float* __restrict__ xi_raw,
                                 const float* __restrict__ conv_w,
                                 const float* __restrict__ conv_b,
                                 float* __restrict__ xi_act) {
    const int d   = threadIdx.x;
    const int row = blockIdx.x;                // b*L + l
    const int l   = row & (LTOT - 1);
    float acc = conv_b[d];
#pragma unroll
    for (int k = 0; k < 4; ++k) {
        const int dl = l - 3 + k;
        const float v = (dl >= 0) ? xi_raw[(size_t)(row - 3 + k) * DIN + d] : 0.0f;
        acc = fmaf(v, conv_w[d * 4 + k], acc);
    }
    xi_act[(size_t)row * DIN + d] = silu_f(acc);
}

// ---------------------------------------------------------------------------
// GEMM-C: [dt_pre | B | C] = xi_act @ Wcat.  Block = 4 waves = 4 M-tiles,
// one 16-col weight tile staged in LDS (16 KB).  A is row-major so each
// lane's K-pair is one 8B load.
// ---------------------------------------------------------------------------
__global__ void gemm_dbl_kernel(const float* __restrict__ xi_act,
                                const float* __restrict__ Wcat,
                                float* __restrict__ dtpre,
                                float* __restrict__ bc) {
    __shared__ float wlds[256 * 16];           // 16 KB
    const int tid  = threadIdx.x;
    const int wave = tid >> 5;
    const int lane = tid & 31;
    const int row0 = blockIdx.x * 64 + wave * 16;
    const int nt   = blockIdx.y;               // 0..16
    const int m    = lane & 15;
    const int kh   = lane >> 4;

    for (int i = tid; i < 256 * 4; i += 128) { // 1024 v4f loads
        const int row = i >> 2;
        const int c4  = (i & 3) * 4;
        *reinterpret_cast<v4f*>(&wlds[row * 16 + c4]) =
            *reinterpret_cast<const v4f*>(&Wcat[(size_t)row * NC_DBL + nt * 16 + c4]);
    }
    __syncthreads();

    v8f acc;
    zero8(acc);
    const float* ab = xi_act + (size_t)(row0 + m) * DIN + 2 * kh;
    for (int k0 = 0; k0 < DIN; k0 += 4) {
        const v2f a  = *reinterpret_cast<const v2f*>(ab + k0);
        const int kl = k0 + 2 * kh;
        v2f bv;
        bv.x = wlds[kl * 16 + m];
        bv.y = wlds[(kl + 1) * 16 + m];
        acc = wmma_f32(a, bv, acc);
    }

    const int mo = kh * 8;
    if (nt < 16) {
#pragma unroll
        for (int r = 0; r < 8; ++r)
            dtpre[(size_t)(row0 + mo + r) * DIN + nt * 16 + m] = acc[r];
    } else if (m < 4) {                        // B0,B1,C0,C1
#pragma unroll
        for (int r = 0; r < 8; ++r)
            bc[(size_t)(row0 + mo + r) * 4 + m] = acc[r];
    }
}

// ---------------------------------------------------------------------------
// Chunked parallel scan (3 passes).  h_l = dA_l * h_{l-1} + inc_l is
// associative -> 32 chunks of 128 steps per batch, 256 workgroups in flight
// instead of 8.  Pass 1: per-chunk (prod dA, h_end).  Pass 2: 32-step carry
// scan per (b,d).  Pass 3: replay with correct h_in, fused output epilogue.
// ---------------------------------------------------------------------------
__global__ void scan_pass1_kernel(const float* __restrict__ xi_act,
                                  const float* __restrict__ dtpre,
                                  const float* __restrict__ bcb,
                                  const float* __restrict__ b_dt,
                                  const float* __restrict__ A_log,
                                  float* __restrict__ ap0a, float* __restrict__ ap1a,
                                  float* __restrict__ he0a, float* __restrict__ he1a) {
    const int blk = blockIdx.x;
    const int b   = blk >> 5;
    const int ch  = blk & (NCHUNK - 1);
    const int d   = threadIdx.x;
    const float bdt = b_dt[d];
    const float A0  = -__expf(A_log[2 * d + 0]);
    const float A1  = -__expf(A_log[2 * d + 1]);

    float ap0 = 1.0f, ap1 = 1.0f, h0 = 0.0f, h1 = 0.0f;
    const size_t base = ((size_t)b * LTOT + ch * CLEN) * DIN + d;
    const size_t bcb0 = ((size_t)b * LTOT + ch * CLEN) * 4;

    for (int l = 0; l < CLEN; ++l) {
        const size_t off = base + (size_t)l * DIN;
        const float xv = xi_act[off];
        const float dp = dtpre[off] + bdt;
        const float B0 = bcb[bcb0 + l * 4 + 0];
        const float B1 = bcb[bcb0 + l * 4 + 1];
        const float dt  = softplus_f(dp);
        const float dA0 = __expf(dt * A0);
        const float dA1 = __expf(dt * A1);
        const float dtx = dt * xv;
        h0 = fmaf(dA0, h0, dtx * B0);
        h1 = fmaf(dA1, h1, dtx * B1);
        ap0 *= dA0;
        ap1 *= dA1;
    }
    const int idx = (b * NCHUNK + ch) * DIN + d;
    ap0a[idx] = ap0; ap1a[idx] = ap1;
    he0a[idx] = h0;  he1a[idx] = h1;
}

__global__ void scan_pass2_kernel(const float* __restrict__ ap0a, const float* __restrict__ ap1a,
                                  const float* __restrict__ he0a, const float* __restrict__ he1a,
                                  float* __restrict__ hi0a, float* __restrict__ hi1a) {
    const int b = blockIdx.x;
    const int d = threadIdx.x;
    float h0 = 0.0f, h1 = 0.0f;
    for (int ch = 0; ch < NCHUNK; ++ch) {
        const int idx = (b * NCHUNK + ch) * DIN + d;
        hi0a[idx] = h0;
        hi1a[idx] = h1;
        h0 = fmaf(ap0a[idx], h0, he0a[idx]);
        h1 = fmaf(ap1a[idx], h1, he1a[idx]);
    }
}

__global__ void scan_pass3_kernel(const float* __restrict__ xi_act,
                                  const float* __restrict__ dtpre,
                                  const float* __restrict__ bcb,
                                  const float* __restrict__ zbuf,
                                  const float* __restrict__ b_dt,
                                  const float* __restrict__ A_log,
                                  const float* __restrict__ Dskip,
                                  const float* __restrict__ hi0a,
                                  const float* __restrict__ hi1a,
                                  float* __restrict__ y) {
    const int blk = blockIdx.x;
    const int b   = blk >> 5;
    const int ch  = blk & (NCHUNK - 1);
    const int d   = threadIdx.x;
    const float bdt = b_dt[d];
    const float dsk = Dskip[d];
    const float A0  = -__expf(A_log[2 * d + 0]);
    const float A1  = -__expf(A_log[2 * d + 1]);

    const int cidx = (b * NCHUNK + ch) * DIN + d;
    float h0 = hi0a[cidx];
    float h1 = hi1a[cidx];

    const size_t base = ((size_t)b * LTOT + ch * CLEN) * DIN + d;
    const size_t bcb0 = ((size_t)b * LTOT + ch * CLEN) * 4;

    for (int l = 0; l < CLEN; ++l) {
        const size_t off = base + (size_t)l * DIN;
        const float xv = xi_act[off];
        const float dp = dtpre[off] + bdt;
        const float zv = zbuf[off];
        const float B0 = bcb[bcb0 + l * 4 + 0];
        const float B1 = bcb[bcb0 + l * 4 + 1];
        const float C0 = bcb[bcb0 + l * 4 + 2];
        const float C1 = bcb[bcb0 + l * 4 + 3];
        const float dt  = softplus_f(dp);
        const float dA0 = __expf(dt * A0);
        const float dA1 = __expf(dt * A1);
        const float dtx = dt * xv;
        h0 = fmaf(dA0, h0, dtx * B0);
        h1 = fmaf(dA1, h1, dtx * B1);
        const float yv = fmaf(h0, C0, h1 * C1) + dsk * xv;
        y[off] = yv * silu_f(zv);
    }
}

// ---------------------------------------------------------------------------
// GEMM-E: out = y @ W_out.  Weight tile LDS-staged; 16x16 D-tile transposed
// through LDS so NCHW (b, oc, L) stores are contiguous float4 runs along L.
// ---------------------------------------------------------------------------
__global__ void gemm_out_kernel(const float* __restrict__ y,
                                const float* __restrict__ Wout,
                                float* __restrict__ out) {
    __shared__ float wlds[256 * 16];           // 16 KB weights
    __shared__ float tlds[4 * 16 * 17];        // per-wave transpose tiles
    const int tid  = threadIdx.x;
    const int wave = tid >> 5;
    const int lane = tid & 31;
    const int row0 = blockIdx.x * 64 + wave * 16;
    const int nt   = blockIdx.y;               // output-channel tile 0..15
    const int m    = lane & 15;
    const int kh   = lane >> 4;

    for (int i = tid; i < 256 * 4; i += 128) {
        const int row = i >> 2;
        const int c4  = (i & 3) * 4;
        *reinterpret_cast<v4f*>(&wlds[row * 16 + c4]) =
            *reinterpret_cast<const v4f*>(&Wout[(size_t)row * DIN + nt * 16 + c4]);
    }
    __syncthreads();

    v8f acc;
    zero8(acc);
    const float* ab = y + (size_t)(row0 + m) * DIN + 2 * kh;
    for (int k0 = 0; k0 < DIN; k0 += 4) {
        const v2f a  = *reinterpret_cast<const v2f*>(ab + k0);
        const int kl = k0 + 2 * kh;
        v2f bv;
        bv.x = wlds[kl * 16 + m];
        bv.y = wlds[(kl + 1) * 16 + m];
        acc = wmma_f32(a, bv, acc);
    }

    // D fragment -> LDS tile (per wave): tl[M][N]
    float* tl = tlds + wave * 16 * 17;
    const int mo = kh * 8;
#pragma unroll
    for (int r = 0; r < 8; ++r)
        tl[(r + mo) * 17 + m] = acc[r];
    __syncthreads();

    const int b   = row0 / LTOT;
    const int l0  = row0 % LTOT;
    const int oc0 = nt * 16;
#pragma unroll
    for (int rep = 0; rep < 2; ++rep) {
        const int oc  = rep * 8 + (lane >> 2);
        const int lch = (lane & 3) * 4;
        v4f vals;
#pragma unroll
        for (int j = 0; j < 4; ++j)
            vals[j] = tl[(lch + j) * 17 + oc];
        *reinterpret_cast<v4f*>(out + (size_t)(b * 256 + oc0 + oc) * LTOT + l0 + lch) = vals;
    }
}

// ---------------------------------------------------------------------------
extern "C" void kernel_launch(void* const* d_in, const int* in_sizes, int n_in,
                              void* d_out, int out_size, void* d_ws, size_t ws_size,
                              hipStream_t stream) {
    const float* x      = (const float*)d_in[0];
    const float* W_in   = (const float*)d_in[1];
    const float* conv_w = (const float*)d_in[2];
    const float* conv_b = (const float*)d_in[3];
    const float* W_x    = (const float*)d_in[4];
    const float* W_dt   = (const float*)d_in[5];
    const float* b_dt   = (const float*)d_in[6];
    const float* A_log  = (const float*)d_in[7];
    const float* Dskip  = (const float*)d_in[8];
    const float* W_out  = (const float*)d_in[9];
    float* out = (float*)d_out;

    float* ws     = (float*)d_ws;
    const size_t NB = (size_t)NROWS * DIN;     // 8,388,608 floats
    float* xi_raw = ws;                        // reused as y (dead after conv)
    float* zbuf   = xi_raw + NB;
    float* xi_act = zbuf + NB;
    float* dtpre  = xi_act + NB;
    float* bcbuf  = dtpre + NB;                // NROWS*4
    float* wcat   = bcbuf + (size_t)NROWS * 4; // 256*272
    float* ap0    = wcat + (size_t)DIN * NC_DBL;
    float* ap1    = ap0 + CHTOT;
    float* he0    = ap1 + CHTOT;
    float* he1    = he0 + CHTOT;
    float* hi0    = he1 + CHTOT;
    float* hi1    = hi0 + CHTOT;
    float* ybuf   = xi_raw;                    // alias

    build_wcat_kernel<<<(DIN * NC_DBL + 255) / 256, 256, 0, stream>>>(W_x, W_dt, wcat);
    gemm_in_kernel<<<dim3(NROWS / 64, 8), 128, 0, stream>>>(x, W_in, xi_raw, zbuf);
    conv_silu_kernel<<<NROWS, DIN, 0, stream>>>(xi_raw, conv_w, conv_b, xi_act);
    gemm_dbl_kernel<<<dim3(NROWS / 64, 17), 128, 0, stream>>>(xi_act, wcat, dtpre, bcbuf);
    scan_pass1_kernel<<<BATCH * NCHUNK, DIN, 0, stream>>>(xi_act, dtpre, bcbuf, b_dt, A_log,
                                                          ap0, ap1, he0, he1);
    scan_pass2_kernel<<<BATCH, DIN, 0, stream>>>(ap0, ap1, he0, he1, hi0, hi1);
    scan_pass3_kernel<<<BATCH * NCHUNK, DIN, 0, stream>>>(xi_act, dtpre, bcbuf, zbuf, b_dt, A_log,
                                                          Dskip, hi0, hi1, ybuf);
    gemm_out_kernel<<<dim3(NROWS / 64, 16), 128, 0, stream>>>(ybuf, W_out, out);
}